// MambaBlock_1176821039916
// MI455X (gfx1250) — compile-verified
//
#include <hip/hip_runtime.h>
#include <hip/hip_bf16.h>
#include <math.h>

// Mamba block forward, fp32, for gfx1250 (MI455X).
// GEMMs: V_WMMA_F32_16X16X4_F32, block = 8 waves, block tile 128(M) x 128(N).
// W strip staged in LDS (shared by all 8 waves), fragments batch-loaded into
// registers then 32 WMMAs issued back-to-back per 16-K subblock.
// Scan: chunked linear-recurrence decomposition (dA[n] = exp(-dt)^(n+1)).
//
// Workspace (floats), ~93.1 MB total.

typedef __attribute__((ext_vector_type(2))) float v2f;
typedef __attribute__((ext_vector_type(8))) float v8f;

namespace {
constexpr int Bsz = 2, Lseq = 2048, DIMC = 1024, NST = 16, KCONV = 4;
constexpr int CHUNK = 64, NCHUNK = Lseq / CHUNK;   // 32 chunks
constexpr int ML   = Bsz * Lseq;                   // 4096 rows
constexpr int NDBL = 2 * NST + DIMC;               // 1056
constexpr int KC   = 32;                           // K-chunk staged in LDS
constexpr int LDST = 36;                           // padded LDS row stride (dwords)
}

static __device__ __forceinline__ float softplusf(float v) {
  return v > 20.f ? v : logf(1.f + __expf(v));
}
static __device__ __forceinline__ float siluf(float v) {
  return v / (1.f + __expf(-v));
}

// ---------------------------------------------------------------------------
// C[M,N] = A[M,K] * W[N,K]^T (row-major), fp32 WMMA 16x16x4.
// Grid: x = ceil(N/128), y = M/128. Block: 256 threads = 8 waves.
// Wave w: rows m0 = (by*8+w)*16; all waves share cols n0b = bx*128.
// ---------------------------------------------------------------------------
__global__ __launch_bounds__(256) void gemm_wmma_f32(
    const float* __restrict__ A, const float* __restrict__ W,
    float* __restrict__ C, int N, int K)
{
  __shared__ float wtile[128 * LDST];   // 18 KB

  const int lane = threadIdx.x & 31;
  const int wave = threadIdx.x >> 5;
  const int m0  = (blockIdx.y * 8 + wave) * 16;
  const int n0b = blockIdx.x * 128;
  const int r  = lane & 15;   // A-row / B-col within 16
  const int hi = lane >> 4;   // k-half for frags, m-half for D

  int nsub = (N - n0b + 15) >> 4;   // wave-uniform
  if (nsub > 8) nsub = 8;

  const float* ap = A + (size_t)(m0 + r) * K + 2 * hi;

  const v8f vzero = {0.f, 0.f, 0.f, 0.f, 0.f, 0.f, 0.f, 0.f};
  v8f acc[8];
#pragma unroll
  for (int j = 0; j < 8; ++j) acc[j] = vzero;

  for (int kc = 0; kc < K; kc += KC) {
    __syncthreads();
    // Cooperative W chunk load: 128 rows x 32 k -> LDS (padded stride 36).
#pragma unroll
    for (int idx = 0; idx < 4; ++idx) {
      const int flat = threadIdx.x + idx * 256;    // 0..1023
      const int row  = flat >> 3;
      const int seg  = (flat & 7) * 4;
      int nn = n0b + row;
      if (nn >= N) nn = N - 1;                     // clamp (valid mem, unused tiles)
      const float4 v = *(const float4*)(W + (size_t)nn * K + kc + seg);
      *(float4*)&wtile[row * LDST + seg] = v;
    }
    __syncthreads();

    // 8 k-steps per chunk, in two 4-step subblocks: batch loads, then WMMAs.
#pragma unroll
    for (int sb = 0; sb < 2; ++sb) {
      v2f a[4];
#pragma unroll
      for (int u = 0; u < 4; ++u)
        a[u] = *(const v2f*)(ap + kc + sb * 16 + 4 * u);

      v2f bf[8][4];
#pragma unroll
      for (int j = 0; j < 8; ++j)
#pragma unroll
        for (int u = 0; u < 4; ++u)
          bf[j][u] = *(const v2f*)&wtile[(j * 16 + r) * LDST + sb * 16 + 4 * u + 2 * hi];

#pragma unroll
      for (int u = 0; u < 4; ++u)
#pragma unroll
        for (int j = 0; j < 8; ++j)
          acc[j] = __builtin_amdgcn_wmma_f32_16x16x4_f32(
              false, a[u], false, bf[j][u], (short)0, acc[j], false, false);
    }
  }

#pragma unroll
  for (int j = 0; j < 8; ++j) {
    if (j < nsub) {   // wave-uniform guard
      const int n = n0b + j * 16 + r;
#pragma unroll
      for (int v = 0; v < 8; ++v)
        C[(size_t)(m0 + v + 8 * hi) * N + n] = acc[j][v];
    }
  }
}

// ---------------------------------------------------------------------------
// Depthwise causal conv (K=4) + bias + SiLU.  Reads x_c half of xz.
// ---------------------------------------------------------------------------
__global__ __launch_bounds__(256) void conv_silu_kernel(
    const float* __restrict__ xz, const float* __restrict__ convw,
    const float* __restrict__ convb, float* __restrict__ xc)
{
  const int idx = blockIdx.x * 256 + threadIdx.x;
  const int d = idx & (DIMC - 1);
  const int l = (idx >> 10) & (Lseq - 1);
  const int b = idx >> 21;

  const float4 w4 = *(const float4*)(convw + d * KCONV);
  const float wv[4] = {w4.x, w4.y, w4.z, w4.w};
  float acc = convb[d];
  const float* base = xz + (size_t)b * Lseq * (2 * DIMC) + d;
#pragma unroll
  for (int k = 0; k < KCONV; ++k) {
    const int li = l - (KCONV - 1) + k;
    if (li >= 0) acc += base[(size_t)li * (2 * DIMC)] * wv[k];
  }
  xc[(size_t)idx] = siluf(acc);
}

// ---------------------------------------------------------------------------
// Scan phase 1: per (b,d,chunk) local recurrence with h0 = 0.
// ---------------------------------------------------------------------------
__global__ __launch_bounds__(256) void scan_local_kernel(
    const float* __restrict__ xdbl, const float* __restrict__ xc,
    float* __restrict__ hend, float* __restrict__ sumdt)
{
  const int d = blockIdx.x * 256 + threadIdx.x;
  const int c = blockIdx.y;
  const int b = blockIdx.z;

  float h[NST];
#pragma unroll
  for (int n = 0; n < NST; ++n) h[n] = 0.f;
  float sdt = 0.f;

  const size_t blbase = (size_t)b * Lseq + (size_t)c * CHUNK;
  for (int t = 0; t < CHUNK; ++t) {
    const float* row = xdbl + (blbase + t) * NDBL;
    float Bv[NST];
    *(float4*)&Bv[0]  = *(const float4*)(row + 0);
    *(float4*)&Bv[4]  = *(const float4*)(row + 4);
    *(float4*)&Bv[8]  = *(const float4*)(row + 8);
    *(float4*)&Bv[12] = *(const float4*)(row + 12);

    const float dt = softplusf(row[2 * NST + d]);
    sdt += dt;
    const float e   = __expf(-dt);              // dA[n] = e^(n+1)
    const float dtx = dt * xc[(blbase + t) * DIMC + d];
    float p = e;
#pragma unroll
    for (int n = 0; n < NST; ++n) { h[n] = h[n] * p + dtx * Bv[n]; p *= e; }
  }

  const size_t o = (((size_t)b * DIMC + d) * NCHUNK + c) * NST;
#pragma unroll
  for (int n = 0; n < NST; n += 4) *(float4*)(hend + o + n) = *(float4*)&h[n];
  sumdt[((size_t)b * DIMC + d) * NCHUNK + c] = sdt;
}

// ---------------------------------------------------------------------------
// Scan phase 2: sequential cross-chunk combine, parallel over b*d*n.
// ---------------------------------------------------------------------------
__global__ __launch_bounds__(256) void scan_combine_kernel(
    const float* __restrict__ hend, const float* __restrict__ sumdt,
    float* __restrict__ hstart)
{
  const int tid = blockIdx.x * 256 + threadIdx.x;
  const int n  = tid & (NST - 1);
  const int bd = tid >> 4;           // b*DIM + d
  float carry = 0.f;
  for (int c = 0; c < NCHUNK; ++c) {
    const size_t ci = (size_t)bd * NCHUNK + c;
    hstart[ci * NST + n] = carry;
    const float a = __expf(-sumdt[ci] * (float)(n + 1));
    carry = hend[ci * NST + n] + a * carry;
  }
}

// ---------------------------------------------------------------------------
// Scan phase 3: re-run chunks with correct h0, fused y = C.h + D*xc, gate.
// ---------------------------------------------------------------------------
__global__ __launch_bounds__(256) void scan_final_kernel(
    const float* __restrict__ xdbl, const float* __restrict__ xc,
    const float* __restrict__ xz, const float* __restrict__ Dp,
    const float* __restrict__ hstart, float* __restrict__ ybuf)
{
  const int d = blockIdx.x * 256 + threadIdx.x;
  const int c = blockIdx.y;
  const int b = blockIdx.z;

  float h[NST];
  const size_t o = (((size_t)b * DIMC + d) * NCHUNK + c) * NST;
#pragma unroll
  for (int n = 0; n < NST; n += 4) *(float4*)&h[n] = *(const float4*)(hstart + o + n);

  const float dcoef = Dp[d];
  const size_t blbase = (size_t)b * Lseq + (size_t)c * CHUNK;
  for (int t = 0; t < CHUNK; ++t) {
    const size_t bl = blbase + t;
    const float* row = xdbl + bl * NDBL;
    float Bv[NST], Cv[NST];
    *(float4*)&Bv[0]  = *(const float4*)(row + 0);
    *(float4*)&Bv[4]  = *(const float4*)(row + 4);
    *(float4*)&Bv[8]  = *(const float4*)(row + 8);
    *(float4*)&Bv[12] = *(const float4*)(row + 12);
    *(float4*)&Cv[0]  = *(const float4*)(row + 16);
    *(float4*)&Cv[4]  = *(const float4*)(row + 20);
    *(float4*)&Cv[8]  = *(const float4*)(row + 24);
    *(float4*)&Cv[12] = *(const float4*)(row + 28);

    const float dt  = softplusf(row[2 * NST + d]);
    const float e   = __expf(-dt);
    const float xcv = xc[bl * DIMC + d];
    const float dtx = dt * xcv;
    float p = e, y = 0.f;
#pragma unroll
    for (int n = 0; n < NST; ++n) {
      h[n] = h[n] * p + dtx * Bv[n];   // h_t = dA*h_{t-1} + dt*B*x
      y += h[n] * Cv[n];               // y_t = C . h_t
      p *= e;
    }
    y += dcoef * xcv;
    const float z = xz[bl * (2 * DIMC) + DIMC + d];
    ybuf[bl * DIMC + d] = y * siluf(z);
  }
}

// ---------------------------------------------------------------------------
extern "C" void kernel_launch(void* const* d_in, const int* in_sizes, int n_in,
                              void* d_out, int out_size, void* d_ws, size_t ws_size,
                              hipStream_t stream)
{
  (void)in_sizes; (void)n_in; (void)out_size; (void)ws_size;
  const float* x     = (const float*)d_in[0];   // (B,L,D)
  const float* Win   = (const float*)d_in[1];   // (2D,D)
  const float* convw = (const float*)d_in[2];   // (D,1,4)
  const float* convb = (const float*)d_in[3];   // (D,)
  const float* Wxp   = (const float*)d_in[4];   // (2N+D, D)
  const float* Dp    = (const float*)d_in[5];   // (D,)
  const float* Wout  = (const float*)d_in[6];   // (D,D)
  float* out = (float*)d_out;

  float* ws   = (float*)d_ws;
  float* xz   = ws;                                       // ML * 2048
  float* xc   = xz   + (size_t)ML * 2 * DIMC;             // ML * 1024
  float* xdbl = xc   + (size_t)ML * DIMC;                 // ML * 1056
  float* ybuf = xdbl + (size_t)ML * NDBL;                 // ML * 1024
  float* hend = ybuf + (size_t)ML * DIMC;                 // 2*1024*32*16
  float* hst  = hend + (size_t)Bsz * DIMC * NCHUNK * NST; // 2*1024*32*16
  float* sdt  = hst  + (size_t)Bsz * DIMC * NCHUNK * NST; // 2*1024*32

  const dim3 blk(256);

  // 1) xz = x @ in_proj_w^T        (4096 x 2048, K=1024)
  gemm_wmma_f32<<<dim3(2 * DIMC / 128, ML / 128), blk, 0, stream>>>(
      x, Win, xz, 2 * DIMC, DIMC);

  // 2) depthwise causal conv + bias + silu -> xc
  conv_silu_kernel<<<(Bsz * Lseq * DIMC) / 256, blk, 0, stream>>>(
      xz, convw, convb, xc);

  // 3) x_dbl = xc @ x_proj_w^T     (4096 x 1056, K=1024)
  gemm_wmma_f32<<<dim3((NDBL + 127) / 128, ML / 128), blk, 0, stream>>>(
      xc, Wxp, xdbl, NDBL, DIMC);

  // 4-6) chunked selective scan
  scan_local_kernel<<<dim3(DIMC / 256, NCHUNK, Bsz), blk, 0, stream>>>(
      xdbl, xc, hend, sdt);
  scan_combine_kernel<<<(Bsz * DIMC * NST) / 256, blk, 0, stream>>>(
      hend, sdt, hst);
  scan_final_kernel<<<dim3(DIMC / 256, NCHUNK, Bsz), blk, 0, stream>>>(
      xdbl, xc, xz, Dp, hst, ybuf);

  // 7) out = y @ out_proj_w^T      (4096 x 1024, K=1024)
  gemm_wmma_f32<<<dim3(DIMC / 128, ML / 128), blk, 0, stream>>>(
      ybuf, Wout, out, DIMC, DIMC);
}